// Attention_layer_64089501991542
// MI455X (gfx1250) — compile-verified
//
#include <hip/hip_runtime.h>

typedef __attribute__((ext_vector_type(2))) float v2f;
typedef __attribute__((ext_vector_type(8))) float v8f;

#define WMMA_F32(a, b, c) \
  __builtin_amdgcn_wmma_f32_16x16x4_f32(false, (a), false, (b), (short)0, (c), false, false)

constexpr int B_ = 16, S_ = 2048, H_ = 768, CA_ = 32, IN_ = 300;
constexpr int SC = 128;             // s-chunk per scores block
constexpr int NCHUNK = S_ / SC;     // 16
constexpr int KSPLIT = 4;           // deterministic K-split for output GEMM
constexpr int KLEN = S_ / KSPLIT;   // 512
constexpr int TQ_STRIDE = 772;      // 768 + 4 pad -> bank-conflict-free A-frag reads

// workspace layout (in floats)
constexpr size_t TQ_OFF  = 0;
constexpr size_t TQ_SZ   = (size_t)B_ * CA_ * H_;            // 393216
constexpr size_t AL_OFF  = TQ_OFF + TQ_SZ;
constexpr size_t AL_SZ   = (size_t)B_ * CA_ * S_;            // 1048576
constexpr size_t DEN_OFF = AL_OFF + AL_SZ;
constexpr size_t DEN_SZ  = (size_t)B_ * NCHUNK * CA_;        // 8192
constexpr size_t NUM_OFF = DEN_OFF + DEN_SZ;
constexpr size_t NUM_SZ  = (size_t)KSPLIT * B_ * CA_ * H_;   // 1572864

// ---------------- Kernel 1: temp_query = tanh(query @ W + b) ----------------
// grid = B*CA blocks, 256 threads; each thread produces 3 h-outputs.
__global__ void k_tq(const float* __restrict__ query, const float* __restrict__ W,
                     const float* __restrict__ bias, float* __restrict__ tq) {
  const int bc = blockIdx.x;                   // b*32 + c
  __shared__ float qs[IN_];
  const float* qrow = query + (size_t)bc * IN_;
  for (int i = threadIdx.x; i < IN_; i += 256) qs[i] = qrow[i];
  __syncthreads();
  const int h0 = threadIdx.x;
  float a0 = bias[h0], a1 = bias[h0 + 256], a2 = bias[h0 + 512];
  for (int i = 0; i < IN_; ++i) {
    const float qv = qs[i];
    const float* wr = W + (size_t)i * H_ + h0;
    a0 = fmaf(qv, wr[0],   a0);
    a1 = fmaf(qv, wr[256], a1);
    a2 = fmaf(qv, wr[512], a2);
  }
  float* o = tq + (size_t)bc * H_ + h0;
  o[0]   = tanhf(a0);
  o[256] = tanhf(a1);
  o[512] = tanhf(a2);
}

// ------- Kernel 2: alpha = exp(scores) * mask, per-chunk denominator -------
// grid = B*NCHUNK blocks, 256 threads (8 waves). Wave w handles s-tile
// [s0+16w, s0+16w+16) x all 32 c via two 16x16 WMMA accumulators, K=768.
__global__ void k_scores(const float* __restrict__ key, const float* __restrict__ tq,
                         const int* __restrict__ seq_len,
                         float* __restrict__ alpha_g, float* __restrict__ den_part) {
  extern __shared__ float smem[];
  float* tqs  = smem;                       // 32 x TQ_STRIDE
  float* dens = smem + 32 * TQ_STRIDE;      // 32 x 8 wave partial denominators

  const int b     = blockIdx.x / NCHUNK;
  const int chunk = blockIdx.x % NCHUNK;
  const int s0    = chunk * SC;
  const int tid   = threadIdx.x;
  const int lane  = tid & 31;
  const int wv    = tid >> 5;               // 0..7
  const int hl    = lane & 15;              // lane % 16
  const int half  = lane >> 4;              // 0/1
  const int kh    = half * 2;               // K sub-offset 0 or 2

  // Stage tq[b] (32x768) into LDS, float4 coalesced, padded rows.
  const float4* tqg = (const float4*)(tq + (size_t)b * CA_ * H_);
  for (int i = tid; i < 32 * (H_ / 4); i += 256) {
    const int row = i / (H_ / 4), c4 = i % (H_ / 4);
    const float4 v = tqg[i];
    float* d = &tqs[row * TQ_STRIDE + c4 * 4];
    d[0] = v.x; d[1] = v.y; d[2] = v.z; d[3] = v.w;
  }
  __syncthreads();

  // A-fragments: tq rows c=0..15 (tile0) and c=16..31 (tile1).
  const float* a0p  = &tqs[hl * TQ_STRIDE + kh];
  const float* a1p  = &tqs[(hl + 16) * TQ_STRIDE + kh];
  // B-fragment: key[s = s0+16w+hl][h = k+kh .. +1] (contiguous pair per lane).
  const float* krow = key + ((size_t)b * S_ + (s0 + wv * 16 + hl)) * H_ + kh;

  v8f acc0 = {0.f, 0.f, 0.f, 0.f, 0.f, 0.f, 0.f, 0.f};
  v8f acc1 = {0.f, 0.f, 0.f, 0.f, 0.f, 0.f, 0.f, 0.f};
  for (int k = 0; k < H_; k += 4) {
    const v2f a0 = *(const v2f*)(a0p + k);
    const v2f a1 = *(const v2f*)(a1p + k);
    const v2f bf = *(const v2f*)(krow + k);
    acc0 = WMMA_F32(a0, bf, acc0);
    acc1 = WMMA_F32(a1, bf, acc1);
  }

  // Epilogue: mask + exp, store alpha, shuffle-reduce denominator per c-row.
  const int s = s0 + wv * 16 + hl;            // this lane's N (column)
  const bool valid = s < seq_len[b];
  float* ag = alpha_g + (size_t)b * CA_ * S_;
  for (int r = 0; r < 8; ++r) {
    const int c0 = r + 8 * half;              // tile0 row; tile1 row = c0+16
    float v0 = valid ? __expf(acc0[r]) : 0.0f;
    float v1 = valid ? __expf(acc1[r]) : 0.0f;
    ag[(size_t)c0 * S_ + s]        = v0;
    ag[(size_t)(c0 + 16) * S_ + s] = v1;
    float r0 = v0, r1 = v1;
    for (int m = 1; m < 16; m <<= 1) {        // reduce within 16-lane half
      r0 += __shfl_xor(r0, m, 32);
      r1 += __shfl_xor(r1, m, 32);
    }
    if (hl == 0) {
      dens[c0 * 8 + wv]        = r0;
      dens[(c0 + 16) * 8 + wv] = r1;
    }
  }
  __syncthreads();
  if (tid < CA_) {
    float d = 0.f;
    for (int w = 0; w < 8; ++w) d += dens[tid * 8 + w];
    den_part[((size_t)b * NCHUNK + chunk) * CA_ + tid] = d;
  }
}

// ---- Kernel 3: num_part[kc][b][c][h] = sum_{s in kc} alpha[c,s]*value[s,h] ----
// grid = B * KSPLIT * 6 blocks, 256 threads. Wave owns one 16-wide h-tile
// (n = nb*8 + wave) x both c-tiles, K loop of KLEN.
__global__ void k_out(const float* __restrict__ value, const float* __restrict__ alpha_g,
                      float* __restrict__ num_part) {
  const int gid = blockIdx.x;
  const int b   = gid / (KSPLIT * 6);
  const int rem = gid % (KSPLIT * 6);
  const int kc  = rem / 6;
  const int nb  = rem % 6;
  const int tid  = threadIdx.x;
  const int lane = tid & 31;
  const int wv   = tid >> 5;
  const int hl   = lane & 15;
  const int half = lane >> 4;
  const int kh   = half * 2;
  const int n    = nb * 8 + wv;               // h-tile 0..47

  // A: alpha rows c=hl (tile0), c=hl+16 (tile1); K = s within this split.
  const float* arow0 = alpha_g + ((size_t)b * CA_ + hl) * S_ + kc * KLEN + kh;
  const float* arow1 = arow0 + (size_t)16 * S_;
  // B: value[s = kc*KLEN + kh + k (+1)][h = 16n + hl]  (.x/.y stride H_).
  const float* vcol  = value + ((size_t)b * S_ + kc * KLEN + kh) * H_ + n * 16 + hl;

  v8f acc0 = {0.f, 0.f, 0.f, 0.f, 0.f, 0.f, 0.f, 0.f};
  v8f acc1 = {0.f, 0.f, 0.f, 0.f, 0.f, 0.f, 0.f, 0.f};
  for (int k = 0; k < KLEN; k += 4) {
    const v2f a0 = *(const v2f*)(arow0 + k);
    const v2f a1 = *(const v2f*)(arow1 + k);
    v2f bf;
    bf.x = vcol[(size_t)k * H_];
    bf.y = vcol[(size_t)(k + 1) * H_];
    acc0 = WMMA_F32(a0, bf, acc0);
    acc1 = WMMA_F32(a1, bf, acc1);
  }
  float* o = num_part + ((size_t)kc * B_ + b) * CA_ * H_;
  for (int r = 0; r < 8; ++r) {
    const int c = r + 8 * half;
    o[(size_t)c * H_ + n * 16 + hl]        = acc0[r];
    o[(size_t)(c + 16) * H_ + n * 16 + hl] = acc1[r];
  }
}

// ---------------- Kernel 4: out = (ordered-sum num) / (ordered-sum den) ----------------
__global__ void k_fin(const float* __restrict__ num_part, const float* __restrict__ den_part,
                      float* __restrict__ out) {
  const int idx = blockIdx.x * 256 + threadIdx.x;   // b*CA*H + c*H + h
  const int bc  = idx / H_;
  const int b   = bc >> 5, c = bc & 31;
  float num = 0.f;
  for (int kc = 0; kc < KSPLIT; ++kc) num += num_part[(size_t)kc * (B_ * CA_ * H_) + idx];
  float den = 0.f;
  for (int ch = 0; ch < NCHUNK; ++ch) den += den_part[((size_t)b * NCHUNK + ch) * CA_ + c];
  out[idx] = num / den;
}

extern "C" void kernel_launch(void* const* d_in, const int* in_sizes, int n_in,
                              void* d_out, int out_size, void* d_ws, size_t ws_size,
                              hipStream_t stream) {
  (void)in_sizes; (void)n_in; (void)out_size; (void)ws_size;
  const float* key     = (const float*)d_in[0];
  const float* value   = (const float*)d_in[1];
  const float* query   = (const float*)d_in[2];
  const int*   seq_len = (const int*)d_in[3];
  const float* W       = (const float*)d_in[4];
  const float* bias    = (const float*)d_in[5];
  float* ws  = (float*)d_ws;
  float* out = (float*)d_out;

  k_tq<<<B_ * CA_, 256, 0, stream>>>(query, W, bias, ws + TQ_OFF);

  const int smem_bytes = (32 * TQ_STRIDE + 32 * 8) * sizeof(float);  // 99,840 B
  k_scores<<<B_ * NCHUNK, 256, smem_bytes, stream>>>(key, ws + TQ_OFF, seq_len,
                                                     ws + AL_OFF, ws + DEN_OFF);

  k_out<<<B_ * KSPLIT * 6, 256, 0, stream>>>(value, ws + AL_OFF, ws + NUM_OFF);

  k_fin<<<(B_ * CA_ * H_) / 256, 256, 0, stream>>>(ws + NUM_OFF, ws + DEN_OFF, out);
}